// Sae_5875515261481
// MI455X (gfx1250) — compile-verified
//
#include <hip/hip_runtime.h>
#include <hip/hip_bf16.h>
#include <stdint.h>

typedef float v2f __attribute__((ext_vector_type(2)));
typedef float v8f __attribute__((ext_vector_type(8)));

#define B_SZ   1024
#define N_INST 2
#define IN_DIM 1024
#define DSAE   16384
#define ROWS   (B_SZ * N_INST)          // 2048

// d_out layout (floats), in reference return order
#define R_SEG      ((size_t)ROWS * IN_DIM)          // 2,097,152
#define OFF_RECON0 ((size_t)0)
#define OFF_RECON1 (R_SEG)
#define OFF_RECON2 (2 * R_SEG)
#define OFF_TOPK0  (3 * R_SEG)
#define OFF_TOPK1  (OFF_TOPK0 + (size_t)ROWS * 1024)
#define OFF_TOPK2  (OFF_TOPK1 + (size_t)ROWS * 4096)
#define OFF_ACTS   (OFF_TOPK2 + (size_t)ROWS * 16384)   // acts_enc_out == topk level 3

#if __has_builtin(__builtin_amdgcn_global_load_async_to_lds_b128) && \
    __has_builtin(__builtin_amdgcn_s_wait_asynccnt)
#define SAE_ASYNC 1
#else
#define SAE_ASYNC 0
#endif

#if SAE_ASYNC
typedef int v4i_vs __attribute__((vector_size(16)));
typedef __attribute__((address_space(1))) v4i_vs* g128_t;   // global (AS1) 16-byte vector ptr
typedef __attribute__((address_space(3))) v4i_vs* l128_t;   // LDS (AS3) 16-byte vector ptr
#endif

// ---------------- Encoder: acts = relu(x @ W_enc^T + b_enc), f32 WMMA 16x16x4 ----
// Block tile 64(M) x 128(N), KC=32, double-buffered LDS, async global->LDS copies.
__global__ __launch_bounds__(256)
void sae_encoder(const float* __restrict__ x, const float* __restrict__ W,
                 const float* __restrict__ b_enc, float* __restrict__ acts)
{
    __shared__ float At[2][64 * 36];    // 9 KB per buffer
    __shared__ float Wt[2][128 * 36];   // 18 KB per buffer  (total 54 KB)

    const int inst = blockIdx.z;
    const int Mt   = blockIdx.y * 64;
    const int Nt   = blockIdx.x * 128;
    const int tid  = threadIdx.x;
    const int wave = tid >> 5;
    const int lane = tid & 31;
    const int lm   = lane & 15;
    const int half = lane >> 4;
    const int Mw   = (wave >> 1) * 16;   // 4 wave-rows of 16
    const int Nw   = (wave & 1) * 64;    // 2 wave-cols of 64

    v8f acc[4];
#pragma unroll
    for (int cg = 0; cg < 4; ++cg) {
        float bias = b_enc[inst * DSAE + Nt + Nw + cg * 16 + lm];
#pragma unroll
        for (int j = 0; j < 8; ++j) acc[cg][j] = bias;
    }

    const float* xbase = x + ((size_t)Mt * N_INST + inst) * IN_DIM;   // row stride N_INST*IN_DIM
    const float* wbase = W + ((size_t)inst * DSAE + Nt) * IN_DIM;     // row stride IN_DIM

    // per-thread staging coordinates (2 A-float4s + 4 W-float4s per thread)
    int rA[2], fA[2], rW[4], fW[4];
#pragma unroll
    for (int it = 0; it < 2; ++it) { int q = tid + it * 256; rA[it] = q >> 3; fA[it] = q & 7; }
#pragma unroll
    for (int it = 0; it < 4; ++it) { int q = tid + it * 256; rW[it] = q >> 3; fW[it] = q & 7; }

#if SAE_ASYNC
#define SAE_COPY16(g, l) \
    __builtin_amdgcn_global_load_async_to_lds_b128((g128_t)(g), (l128_t)(l), 0, 0)
#else
#define SAE_COPY16(g, l) (*(float4*)(l) = *(const float4*)(g))
#endif

#define SAE_ISSUE(buf, k0)                                                                 \
    do {                                                                                   \
        _Pragma("unroll")                                                                  \
        for (int it = 0; it < 2; ++it) {                                                   \
            const float* g = xbase + (size_t)rA[it] * (N_INST * IN_DIM) + (k0) + fA[it]*4; \
            float* l = &At[buf][rA[it] * 36 + fA[it] * 4];                                 \
            SAE_COPY16(g, l);                                                              \
        }                                                                                  \
        _Pragma("unroll")                                                                  \
        for (int it = 0; it < 4; ++it) {                                                   \
            const float* g = wbase + (size_t)rW[it] * IN_DIM + (k0) + fW[it] * 4;          \
            float* l = &Wt[buf][rW[it] * 36 + fW[it] * 4];                                 \
            SAE_COPY16(g, l);                                                              \
        }                                                                                  \
    } while (0)

    SAE_ISSUE(0, 0);

    for (int i = 0; i < IN_DIM / 32; ++i) {
        const int cur = i & 1;
        const int k0n = (i + 1) * 32;
#if SAE_ASYNC
        if (k0n < IN_DIM) {
            SAE_ISSUE(cur ^ 1, k0n);
            __builtin_amdgcn_s_wait_asynccnt(6);   // current buffer's 6 copies done
        } else {
            __builtin_amdgcn_s_wait_asynccnt(0);
        }
#else
        if (k0n < IN_DIM) SAE_ISSUE(cur ^ 1, k0n);
#endif
        __syncthreads();

#pragma unroll
        for (int kk = 0; kk < 32; kk += 4) {
            const int kc = kk + 2 * half;
            // A frag: lanes 0-15 -> M=lm K={kk,kk+1}; lanes 16-31 -> K={kk+2,kk+3}
            v2f a = *(const v2f*)(&At[cur][(Mw + lm) * 36 + kc]);
#pragma unroll
            for (int cg = 0; cg < 4; ++cg) {
                v2f b = *(const v2f*)(&Wt[cur][(Nw + cg * 16 + lm) * 36 + kc]);
                acc[cg] = __builtin_amdgcn_wmma_f32_16x16x4_f32(
                    false, a, false, b, (short)0, acc[cg], false, false);
            }
        }
        __syncthreads();   // all waves done reading `cur` before it is refilled
    }

    // C layout: VGPR r -> (M = r + 8*half, N = lm), relu + store
#pragma unroll
    for (int cg = 0; cg < 4; ++cg) {
#pragma unroll
        for (int r = 0; r < 8; ++r) {
            int m = Mt + Mw + r + 8 * half;
            int n = Nt + Nw + cg * 16 + lm;
            float v = acc[cg][r];
            acts[((size_t)m * N_INST + inst) * DSAE + n] = v > 0.f ? v : 0.f;
        }
    }
#undef SAE_COPY16
#undef SAE_ISSUE
}

// ---------------- Global top-k threshold search (3-pass bit histogram) ----------
// ws layout (uint32): [0..2047] histogram ; [2048 + lev*8 + {0:value,1:kRem,2:thr,3:tieAllowed,4:tieCnt}]

__global__ void sae_init_state(unsigned* ws)
{
    if (threadIdx.x == 0 && blockIdx.x == 0) {
        const unsigned ks[3] = {32u * ROWS, 64u * ROWS, 128u * ROWS};
        for (int lev = 0; lev < 3; ++lev) {
            unsigned* st = ws + 2048 + lev * 8;
            st[0] = 0; st[1] = ks[lev]; st[2] = 0; st[3] = 0; st[4] = 0;
        }
    }
}

__global__ void sae_zero_hist(unsigned* ws)
{
    int i = blockIdx.x * 256 + threadIdx.x;
    if (i < 2048) ws[i] = 0;
}

__global__ __launch_bounds__(256)
void sae_hist(const float* __restrict__ acts, unsigned* __restrict__ ws, int pass, int lev)
{
    __shared__ unsigned h[2048];
    for (int i = threadIdx.x; i < 2048; i += 256) h[i] = 0;
    __syncthreads();

    const unsigned vs = ws[2048 + lev * 8 + 0];
    const float* rbase = acts + (size_t)blockIdx.y * DSAE;
    const int s0 = blockIdx.x * 1024;
    for (int s = s0 + threadIdx.x; s < s0 + 1024; s += 256) {
        unsigned u = __float_as_uint(rbase[s]);   // relu'd -> non-negative, uint order == float order
        bool ok; unsigned bin;
        if (pass == 0)      { ok = true;                       bin = u >> 21; }
        else if (pass == 1) { ok = (u >> 21) == (vs >> 21);    bin = (u >> 10) & 0x7FFu; }
        else                { ok = (u >> 10) == (vs >> 10);    bin = u & 0x3FFu; }
        if (ok) atomicAdd(&h[bin], 1u);
    }
    __syncthreads();
    for (int i = threadIdx.x; i < 2048; i += 256)
        if (h[i]) atomicAdd(&ws[i], h[i]);
}

__global__ void sae_reduce(unsigned* ws, int pass, int lev)
{
    if (threadIdx.x != 0 || blockIdx.x != 0) return;
    unsigned* st = ws + 2048 + lev * 8;
    unsigned kRem = st[1];
    int nb = (pass == 2) ? 1024 : 2048;
    unsigned cum = 0, above = 0;
    int binSel = 0;
    for (int b = nb - 1; b >= 0; --b) {
        unsigned c = ws[b];
        if (cum + c >= kRem) { binSel = b; above = cum; break; }
        cum += c;
        above = cum;
    }
    unsigned newRem = (kRem > above) ? (kRem - above) : 0u;
    unsigned value = st[0];
    if (pass == 0)      value = ((unsigned)binSel) << 21;
    else if (pass == 1) value |= ((unsigned)binSel) << 10;
    else                value |= (unsigned)binSel;
    st[0] = value;
    st[1] = newRem;
    if (pass == 2) { st[2] = value; st[3] = newRem; st[4] = 0; }
}

// ---------------- Select/scatter: write full topk segment (zeros elsewhere) -----
__global__ __launch_bounds__(256)
void sae_select(const float* __restrict__ acts, float* __restrict__ outTopk,
                float* __restrict__ actsInplace, unsigned* ws, int d_l, int lev)
{
    unsigned* st = ws + 2048 + lev * 8;
    const unsigned thr = st[2];
    const unsigned tieAllowed = st[3];
    const int row = blockIdx.y;
    const int s = blockIdx.x * 256 + threadIdx.x;
    if (s >= d_l) return;
    float v = acts[(size_t)row * DSAE + s];
    unsigned u = __float_as_uint(v);
    bool keep = u > thr;
    if (u == thr) keep = atomicAdd(&st[4], 1u) < tieAllowed;
    float o = keep ? v : 0.f;
    outTopk[(size_t)row * d_l + s] = o;
    if (actsInplace) actsInplace[(size_t)row * DSAE + s] = o;   // level-3 in-place -> acts_enc_out
}

// ---------------- Sparse decode: recon = relu(topk @ W_dec + b_dec) -------------
// W_dec[i,d,s] == W_enc[i,s,d]  => gather contiguous W_enc rows (coalesced).
__global__ __launch_bounds__(256)
void sae_decode(const float* __restrict__ topk, const float* __restrict__ W,
                const float* __restrict__ b_dec, float* __restrict__ recon, int d_l)
{
    __shared__ int   cnt;
    __shared__ int   sidx[4096];
    __shared__ float sval[4096];

    const int rid = blockIdx.x;            // rid = b*N_INST + i
    const int i = rid & (N_INST - 1);
    const float* trow = topk + (size_t)rid * d_l;

    float acc[4] = {0.f, 0.f, 0.f, 0.f};
    for (int seg = 0; seg < d_l; seg += 4096) {
        if (threadIdx.x == 0) cnt = 0;
        __syncthreads();
        const int send = (seg + 4096 < d_l) ? seg + 4096 : d_l;
        for (int s = seg + threadIdx.x; s < send; s += 256) {
            float v = trow[s];
            if (v != 0.f) { int p = atomicAdd(&cnt, 1); sidx[p] = s; sval[p] = v; }
        }
        __syncthreads();
        const int n = cnt;
        for (int e = 0; e < n; ++e) {
            const float* wrow = W + ((size_t)i * DSAE + sidx[e]) * IN_DIM;
            const float v = sval[e];
#pragma unroll
            for (int q = 0; q < 4; ++q) acc[q] += v * wrow[threadIdx.x + q * 256];
        }
        __syncthreads();
    }
#pragma unroll
    for (int q = 0; q < 4; ++q) {
        const int d = threadIdx.x + q * 256;
        float r = acc[q] + b_dec[i * IN_DIM + d];
        recon[(size_t)rid*IN_DIM + d] = r > 0.f ? r : 0.f;
    }
}

// -------------------------------------------------------------------------------
extern "C" void kernel_launch(void* const* d_in, const int* in_sizes, int n_in,
                              void* d_out, int out_size, void* d_ws, size_t ws_size,
                              hipStream_t stream)
{
    (void)in_sizes; (void)n_in; (void)out_size; (void)ws_size;
    const float* x     = (const float*)d_in[0];
    const float* W_enc = (const float*)d_in[1];
    // d_in[2] = W_dec, unused: W_dec[i,d,s] == W_enc[i,s,d]
    const float* b_enc = (const float*)d_in[3];
    const float* b_dec = (const float*)d_in[4];
    float* out = (float*)d_out;
    unsigned* ws = (unsigned*)d_ws;

    float* acts = out + OFF_ACTS;   // dense encoder output; later rewritten in-place to topk3

    // 1) encoder GEMM (f32 WMMA, async double-buffered LDS)
    sae_encoder<<<dim3(DSAE / 128, B_SZ / 64, N_INST), 256, 0, stream>>>(x, W_enc, b_enc, acts);

    // 2) per-level global top-k thresholds
    sae_init_state<<<1, 1, 0, stream>>>(ws);
    const int dls[3] = {1024, 4096, 16384};
    for (int lev = 0; lev < 3; ++lev) {
        for (int pass = 0; pass < 3; ++pass) {
            sae_zero_hist<<<8, 256, 0, stream>>>(ws);
            sae_hist<<<dim3(dls[lev] / 1024, ROWS), 256, 0, stream>>>(acts, ws, pass, lev);
            sae_reduce<<<1, 1, 0, stream>>>(ws, pass, lev);
        }
    }

    // 3) scatter topk (levels 0,1 first; level 2 rewrites acts in place => acts_enc_out)
    sae_select<<<dim3(1024 / 256, ROWS), 256, 0, stream>>>(acts, out + OFF_TOPK0, nullptr, ws, 1024, 0);
    sae_select<<<dim3(4096 / 256, ROWS), 256, 0, stream>>>(acts, out + OFF_TOPK1, nullptr, ws, 4096, 1);
    sae_select<<<dim3(16384 / 256, ROWS), 256, 0, stream>>>(acts, out + OFF_TOPK2, acts, ws, 16384, 2);

    // 4) sparse decode per level
    sae_decode<<<ROWS, 256, 0, stream>>>(out + OFF_TOPK0, W_enc, b_dec, out + OFF_RECON0, 1024);
    sae_decode<<<ROWS, 256, 0, stream>>>(out + OFF_TOPK1, W_enc, b_dec, out + OFF_RECON1, 4096);
    sae_decode<<<ROWS, 256, 0, stream>>>(out + OFF_TOPK2, W_enc, b_dec, out + OFF_RECON2, 16384);
}